// StormCellDetector_11682311045870
// MI455X (gfx1250) — compile-verified
//
#include <hip/hip_runtime.h>

// ---- problem constants (from reference) ----
#define BATCH 32
#define CHAN  128
#define HF    96
#define WF    96
#define HV    384
#define WV    384
#define NFRM  12
#define MAXS  50      // MAX_STORMS
#define MPAD  64      // MAXS padded to 4 WMMA m-tiles
#define MAXC  4096    // candidate cap per batch
#define TILE  32      // peak-detect output tile
#define IN_T  39      // TILE + 7 halo (lo=4, hi=3)

typedef __attribute__((ext_vector_type(2))) float v2f;
typedef __attribute__((ext_vector_type(8))) float v8f;

static __device__ __forceinline__ int imin(int a, int b) { return a < b ? a : b; }
static __device__ __forceinline__ int imax(int a, int b) { return a > b ? a : b; }

// ---------------- kernel 0: zero candidate counters ----------------
__global__ void storm_zero_kernel(int* __restrict__ cnt) {
    int t = threadIdx.x;
    if (t < BATCH) cnt[t] = 0;
}

// ---------------- kernel 1: peak detection (separable 8x8 max filter in LDS) ----------------
// symmetric padding of a contiguous window with half-width < H is equivalent to the
// window clamped to the array, which clamped-index loads reproduce exactly.
__global__ void storm_peak_kernel(const float* __restrict__ vil_input,
                                  int* __restrict__ cnt,
                                  float* __restrict__ cval,
                                  int* __restrict__ cidx) {
    __shared__ float sIn[IN_T][IN_T + 1];  // 39 x 40
    __shared__ float sRM[IN_T][TILE];      // 39 x 32 row-maxes

    const int b   = blockIdx.z;
    const int ty0 = blockIdx.y * TILE;
    const int tx0 = blockIdx.x * TILE;
    const float* base = vil_input + ((size_t)b * NFRM + (NFRM - 1)) * (size_t)(HV * WV);

    for (int e = threadIdx.x; e < IN_T * IN_T; e += blockDim.x) {
        int i = e / IN_T, j = e % IN_T;
        int gy = imin(imax(ty0 - 4 + i, 0), HV - 1);
        int gx = imin(imax(tx0 - 4 + j, 0), WV - 1);
        sIn[i][j] = base[gy * WV + gx];
    }
    __syncthreads();

    // row max over window x-4..x+3  (local cols j..j+7)
    for (int e = threadIdx.x; e < IN_T * TILE; e += blockDim.x) {
        int i = e / TILE, j = e % TILE;
        float m = sIn[i][j];
#pragma unroll
        for (int k = 1; k < 8; ++k) m = fmaxf(m, sIn[i][j + k]);
        sRM[i][j] = m;
    }
    __syncthreads();

    // col max over window y-4..y+3 (local rows yy..yy+7) + peak test
    for (int e = threadIdx.x; e < TILE * TILE; e += blockDim.x) {
        int yy = e / TILE, xx = e % TILE;
        float v = sIn[yy + 4][xx + 4];
        float m = sRM[yy][xx];
#pragma unroll
        for (int k = 1; k < 8; ++k) m = fmaxf(m, sRM[yy + k][xx]);
        if (v > 0.3f && v >= m) {
            int pos = atomicAdd(&cnt[b], 1);
            if (pos < MAXC) {
                cval[b * MAXC + pos] = v;
                cidx[b * MAXC + pos] = (ty0 + yy) * WV + (tx0 + xx);
            }
        }
    }
}

// ---------------- kernel 2: per-batch top-50 (value desc, index asc ties) ----------------
__global__ void storm_topk_kernel(const int* __restrict__ cnt,
                                  const float* __restrict__ cval,
                                  const int* __restrict__ cidx,
                                  int* __restrict__ toplin,
                                  float* __restrict__ validw,
                                  float* __restrict__ out_pos,
                                  float* __restrict__ out_valid) {
    __shared__ float sV[MAXC];
    __shared__ int   sI[MAXC];
    __shared__ float sRv[256];
    __shared__ int   sRi[256];
    __shared__ int   sRp[256];
    __shared__ float sTv[MAXS];
    __shared__ int   sTi[MAXS];

    const int b = blockIdx.x;
    const int t = threadIdx.x;
    const int n = imin(cnt[b], MAXC);

    for (int e = t; e < n; e += 256) { sV[e] = cval[b * MAXC + e]; sI[e] = cidx[b * MAXC + e]; }
    __syncthreads();

    for (int it = 0; it < MAXS; ++it) {
        float bv = -3.4e38f; int bi = 0x7fffffff; int bp = -1;
        for (int e = t; e < n; e += 256) {
            float v = sV[e]; int i = sI[e];
            if (v > bv || (v == bv && i < bi)) { bv = v; bi = i; bp = e; }
        }
        sRv[t] = bv; sRi[t] = bi; sRp[t] = bp;
        __syncthreads();
        for (int s = 128; s > 0; s >>= 1) {
            if (t < s) {
                float v = sRv[t + s]; int i = sRi[t + s];
                if (v > sRv[t] || (v == sRv[t] && i < sRi[t])) {
                    sRv[t] = v; sRi[t] = i; sRp[t] = sRp[t + s];
                }
            }
            __syncthreads();
        }
        if (t == 0) {
            sTv[it] = sRv[0]; sTi[it] = sRi[0];
            if (sRp[0] >= 0) sV[sRp[0]] = -3.4e38f;  // remove winner
        }
        __syncthreads();
    }

    if (t < MAXS) {
        float v = sTv[t];
        int idx = sTi[t];
        bool valid = (v > -1.0e30f);
        if (n == 0 && t == 0) {            // no-peak fallback: center, valid
            idx = (HV / 2) * WV + (WV / 2);
            valid = true;
        }
        if (!valid) idx = 0;
        int y = idx / WV, x = idx % WV;
        int yf = imin(y * HF / HV, HF - 1);
        int xf = imin(x * WF / WV, WF - 1);
        toplin[b * MPAD + t] = yf * WF + xf;
        float vf = valid ? 1.0f : 0.0f;
        validw[b * MPAD + t] = vf;
        out_pos[((size_t)b * MAXS + t) * 2 + 0] = (float)yf;
        out_pos[((size_t)b * MAXS + t) * 2 + 1] = (float)xf;
        out_valid[(size_t)b * MAXS + t] = vf;
    }
}

// ---------------- kernel 3: gather + f32 WMMA projection ----------------
// out[b,m,d] = (sum_c features[b,c,lin(m)] * proj_w[d,c] + proj_b[d]) * valid[b,m]
// A tile (16x4, MxK=c): lane L holds M=L%16, elems K=(L/16)*2+{0,1}  -> float2 from LDS
// B tile (4x16, KxN=d): lane L holds N=L%16, elems K=(L/16)*2+{0,1}  -> float2 from proj_w row-major
// C/D (16x16 f32): vgpr r holds M = r + 8*(L/16), N = L%16
__global__ void storm_proj_kernel(const float* __restrict__ features,
                                  const float* __restrict__ proj_w,
                                  const float* __restrict__ proj_b,
                                  const int* __restrict__ toplin,
                                  const float* __restrict__ validw,
                                  float* __restrict__ out_nf) {
    __shared__ float sG[MPAD][132];   // gathered [m][c], padded stride (33 KB)
    __shared__ int   sLin[MPAD];
    __shared__ float sVf[MPAD];

    const int b = blockIdx.x;
    const int t = threadIdx.x;

    if (t < MPAD) {
        sLin[t] = (t < MAXS) ? toplin[b * MPAD + t] : 0;
        sVf[t]  = (t < MAXS) ? validw[b * MPAD + t] : 0.0f;
    }
    __syncthreads();

    const float* fb = features + (size_t)b * CHAN * (HF * WF);
    for (int e = t; e < MPAD * CHAN; e += 256) {
        int m = e >> 7, c = e & 127;
        sG[m][c] = (m < MAXS) ? fb[(size_t)c * (HF * WF) + sLin[m]] : 0.0f;
    }
    __syncthreads();

    const int w     = t >> 5;        // wave id = d-tile (8 tiles of 16 -> d 0..127)
    const int L     = t & 31;
    const int nlo   = L & 15;
    const int khalf = L >> 4;

    const float* wrow = proj_w + (size_t)(w * 16 + nlo) * CHAN + khalf * 2;

    v8f acc[4] = {};                 // 4 m-tiles of 16
    for (int k0 = 0; k0 < CHAN; k0 += 4) {
        v2f bf = *(const v2f*)(wrow + k0);
#pragma unroll
        for (int mt = 0; mt < 4; ++mt) {
            v2f af = *(const v2f*)&sG[mt * 16 + nlo][k0 + khalf * 2];
            acc[mt] = __builtin_amdgcn_wmma_f32_16x16x4_f32(
                false, af, false, bf, (short)0, acc[mt], false, false);
        }
    }

    const int d = w * 16 + nlo;
    const float bias = proj_b[d];
#pragma unroll
    for (int mt = 0; mt < 4; ++mt) {
#pragma unroll
        for (int r = 0; r < 8; ++r) {
            int m = mt * 16 + r + 8 * khalf;
            if (m < MAXS) {
                out_nf[((size_t)b * MAXS + m) * CHAN + d] = (acc[mt][r] + bias) * sVf[m];
            }
        }
    }
}

// ---------------- launcher ----------------
extern "C" void kernel_launch(void* const* d_in, const int* in_sizes, int n_in,
                              void* d_out, int out_size, void* d_ws, size_t ws_size,
                              hipStream_t stream) {
    const float* features = (const float*)d_in[0];   // (32,128,96,96)
    const float* vil      = (const float*)d_in[1];   // (32,12,384,384)
    const float* proj_w   = (const float*)d_in[2];   // (128,128)
    const float* proj_b   = (const float*)d_in[3];   // (128,)

    float* out       = (float*)d_out;
    float* out_nf    = out;                                   // 32*50*128
    float* out_pos   = out + (size_t)BATCH * MAXS * CHAN;     // 32*50*2
    float* out_valid = out_pos + (size_t)BATCH * MAXS * 2;    // 32*50

    // workspace layout (~1.05 MB)
    int*   cnt    = (int*)d_ws;                               // 32
    int*   toplin = cnt + BATCH;                              // 32*64
    float* validw = (float*)(toplin + BATCH * MPAD);          // 32*64
    float* cval   = validw + BATCH * MPAD;                    // 32*4096
    int*   cidx   = (int*)(cval + BATCH * MAXC);              // 32*4096

    storm_zero_kernel<<<1, 64, 0, stream>>>(cnt);

    dim3 pg(WV / TILE, HV / TILE, BATCH);                     // 12 x 12 x 32
    storm_peak_kernel<<<pg, 256, 0, stream>>>(vil, cnt, cval, cidx);

    storm_topk_kernel<<<BATCH, 256, 0, stream>>>(cnt, cval, cidx,
                                                 toplin, validw, out_pos, out_valid);

    storm_proj_kernel<<<BATCH, 256, 0, stream>>>(features, proj_w, proj_b,
                                                 toplin, validw, out_nf);
}